// QGCN_22239340659451
// MI455X (gfx1250) — compile-verified
//
#include <hip/hip_runtime.h>
#include <hip/hip_bf16.h>

// Problem constants (match reference)
#define NN 100000
#define EE 1600000
#define DD 128
#define LL 4
#define GG 128
#define TT 10
#define EPS_BN 1e-5f

typedef __attribute__((ext_vector_type(2))) float v2f;
typedef __attribute__((ext_vector_type(8))) float v8f;

// ---------------------------------------------------------------------------
// Dense GEMM: Out[N,128] = A[N,128] @ W[128,128] (+ bias) using f32 WMMA.
// One wave computes a 16-row x 128-col strip: 8 accumulators of 16x16,
// K chained through 32 steps of v_wmma_f32_16x16x4_f32.
// A 16x4 layout: lanes 0-15 hold row M=lane {K, K+1}; lanes 16-31 hold {K+2, K+3}.
// B 4x16 layout mirrors A. C/D: VGPR j -> M=j (lanes 0-15) / M=j+8 (lanes 16-31).
// ---------------------------------------------------------------------------
__global__ __launch_bounds__(256) void qgcn_gemm_wmma(
    const float* __restrict__ A, const float* __restrict__ W,
    const float* __restrict__ bias, float* __restrict__ Out, int ntiles)
{
    const int wtile = (int)((blockIdx.x * blockDim.x + threadIdx.x) >> 5); // row tile
    const int lane  = (int)(threadIdx.x & 31);
    if (wtile >= ntiles) return;                 // wave-uniform -> EXEC all-ones inside

    const int lmod  = lane & 15;
    const int lhalf = lane >> 4;                 // 0 or 1
    const int row16 = wtile * 16;

    v8f acc[8] = {};
    const float* arow = A + (size_t)(row16 + lmod) * DD;

    for (int kb = 0; kb < 32; ++kb) {
        const int kk = kb * 4 + 2 * lhalf;
        v2f a;
        a.x = arow[kk];
        a.y = arow[kk + 1];
        #pragma unroll
        for (int j = 0; j < 8; ++j) {
            const int col = j * 16 + lmod;
            v2f b;
            b.x = W[(size_t)kk * DD + col];
            b.y = W[(size_t)(kk + 1) * DD + col];
            acc[j] = __builtin_amdgcn_wmma_f32_16x16x4_f32(
                false, a, false, b, (short)0, acc[j], false, false);
        }
    }

    #pragma unroll
    for (int j = 0; j < 8; ++j) {
        const int col = j * 16 + lmod;
        const float bj = bias ? bias[col] : 0.0f;
        #pragma unroll
        for (int r = 0; r < 8; ++r) {
            const int row = row16 + r + 8 * lhalf;
            Out[(size_t)row * DD + col] = acc[j][r] + bj;
        }
    }
}

// ---------------------------------------------------------------------------
// Degree / normalization
// ---------------------------------------------------------------------------
__global__ void qgcn_deg_init(float* __restrict__ deg, int n) {
    int i = blockIdx.x * blockDim.x + threadIdx.x;
    if (i < n) deg[i] = 1.0f;                    // self-loop
}
__global__ void qgcn_deg_edges(const int* __restrict__ dst, float* __restrict__ deg, int e) {
    int i = blockIdx.x * blockDim.x + threadIdx.x;
    if (i < e) atomicAdd(&deg[dst[i]], 1.0f);
}
__global__ void qgcn_deg_finalize(float* __restrict__ deg, int n) {
    int i = blockIdx.x * blockDim.x + threadIdx.x;
    if (i < n) deg[i] = rsqrtf(fmaxf(deg[i], 1.0f));
}

// ---------------------------------------------------------------------------
// agg[n,d] = bc[d] + hw[n,d] * dinv[n]^2   (self-loop message + bias init)
// ---------------------------------------------------------------------------
__global__ void qgcn_selfloop_init(const float* __restrict__ hw,
                                   const float* __restrict__ dinv,
                                   const float* __restrict__ bc,
                                   float* __restrict__ agg, int total)
{
    int i = blockIdx.x * blockDim.x + threadIdx.x;
    if (i >= total) return;
    int d = i & (DD - 1);
    int n = i >> 7;
    float w = dinv[n];
    agg[i] = bc[d] + hw[i] * w * w;
}

// ---------------------------------------------------------------------------
// Edge scatter: one wave per edge; lane handles 4 features (float4 gather,
// 4 packed f32 atomics into agg[dst]). All traffic is L2-resident.
// ---------------------------------------------------------------------------
__global__ __launch_bounds__(256) void qgcn_edge_scatter(
    const int* __restrict__ src, const int* __restrict__ dst,
    const float* __restrict__ dinv, const float* __restrict__ hw,
    float* __restrict__ agg, int e)
{
    const int gw   = (int)((blockIdx.x * blockDim.x + threadIdx.x) >> 5);
    const int lane = (int)(threadIdx.x & 31);
    if (gw >= e) return;
    const int s = src[gw], d = dst[gw];
    const float nrm = dinv[s] * dinv[d];
    const float4 m = ((const float4*)(hw + (size_t)s * DD))[lane];
    float* ad = agg + (size_t)d * DD + lane * 4;
    atomicAdd(ad + 0, m.x * nrm);
    atomicAdd(ad + 1, m.y * nrm);
    atomicAdd(ad + 2, m.z * nrm);
    atomicAdd(ad + 3, m.w * nrm);
}

// ---------------------------------------------------------------------------
// BatchNorm statistics: per-column sum / sum-of-squares over N rows.
// ---------------------------------------------------------------------------
__global__ void qgcn_zero(float* __restrict__ p, int n) {
    int i = blockIdx.x * blockDim.x + threadIdx.x;
    if (i < n) p[i] = 0.0f;
}
__global__ __launch_bounds__(256) void qgcn_bn_stats(
    const float* __restrict__ agg, float* __restrict__ colsum,
    float* __restrict__ colsq, int nrows)
{
    __shared__ float s_sum[256], s_sq[256];
    const int col  = (int)(threadIdx.x & (DD - 1));
    const int half = (int)(threadIdx.x >> 7);
    float sum = 0.0f, sq = 0.0f;
    for (int r = blockIdx.x * 2 + half; r < nrows; r += gridDim.x * 2) {
        float v = agg[(size_t)r * DD + col];
        sum += v; sq += v * v;
    }
    s_sum[threadIdx.x] = sum; s_sq[threadIdx.x] = sq;
    __syncthreads();
    if (half == 0) {
        atomicAdd(&colsum[col], s_sum[col] + s_sum[col + 128]);
        atomicAdd(&colsq[col],  s_sq[col]  + s_sq[col + 128]);
    }
}
__global__ void qgcn_bn_finalize(const float* __restrict__ colsum,
                                 const float* __restrict__ colsq,
                                 const float* __restrict__ gamma,
                                 float* __restrict__ mu, float* __restrict__ scale)
{
    int d = threadIdx.x;
    if (d < DD) {
        float m = colsum[d] * (1.0f / (float)NN);
        float var = colsq[d] * (1.0f / (float)NN) - m * m;   // biased var (ddof=0)
        mu[d] = m;
        scale[d] = rsqrtf(var + EPS_BN) * gamma[d];
    }
}

// h[n,d] += relu((agg[n,d]-mu[d])*scale[d] + beta[d])   (residual)
__global__ void qgcn_bn_update(float* __restrict__ h, const float* __restrict__ agg,
                               const float* __restrict__ mu, const float* __restrict__ scale,
                               const float* __restrict__ beta, int total)
{
    int i = blockIdx.x * blockDim.x + threadIdx.x;
    if (i >= total) return;
    int d = i & (DD - 1);
    float v = (agg[i] - mu[d]) * scale[d] + beta[d];
    h[i] += fmaxf(v, 0.0f);
}

// ---------------------------------------------------------------------------
// Global mean pool: wave per node; lane 0 counts, each lane adds 4 features.
// ---------------------------------------------------------------------------
__global__ __launch_bounds__(256) void qgcn_pool_sum(
    const float* __restrict__ h, const int* __restrict__ batch,
    float* __restrict__ pooled, float* __restrict__ cnt, int n)
{
    const int gw   = (int)((blockIdx.x * blockDim.x + threadIdx.x) >> 5);
    const int lane = (int)(threadIdx.x & 31);
    if (gw >= n) return;
    const int b = batch[gw];
    if (lane == 0) atomicAdd(&cnt[b], 1.0f);
    const float4 v = ((const float4*)(h + (size_t)gw * DD))[lane];
    float* p = pooled + (size_t)b * DD + lane * 4;
    atomicAdd(p + 0, v.x);
    atomicAdd(p + 1, v.y);
    atomicAdd(p + 2, v.z);
    atomicAdd(p + 3, v.w);
}

// ---------------------------------------------------------------------------
// Readout MLP: one block per graph (G=128). 128 -> 64 -> 32 -> 10.
// ---------------------------------------------------------------------------
__global__ __launch_bounds__(128) void qgcn_mlp(
    const float* __restrict__ pooled, const float* __restrict__ cnt,
    const float* __restrict__ W1, const float* __restrict__ b1,
    const float* __restrict__ W2, const float* __restrict__ b2,
    const float* __restrict__ W3, const float* __restrict__ b3,
    float* __restrict__ out)
{
    __shared__ float p[DD], z1[64], z2[32];
    const int g = blockIdx.x;
    const int t = threadIdx.x;
    const float c = fmaxf(cnt[g], 1.0f);
    p[t] = pooled[(size_t)g * DD + t] / c;
    __syncthreads();
    if (t < 64) {
        float acc = b1[t];
        for (int d = 0; d < 128; ++d) acc += p[d] * W1[d * 64 + t];
        z1[t] = fmaxf(acc, 0.0f);
    }
    __syncthreads();
    if (t < 32) {
        float acc = b2[t];
        for (int d = 0; d < 64; ++d) acc += z1[d] * W2[d * 32 + t];
        z2[t] = fmaxf(acc, 0.0f);
    }
    __syncthreads();
    if (t < TT) {
        float acc = b3[t];
        for (int d = 0; d < 32; ++d) acc += z2[d] * W3[d * TT + t];
        out[(size_t)g * TT + t] = acc;
    }
}

// ---------------------------------------------------------------------------
// Host-side launch orchestration
// ---------------------------------------------------------------------------
extern "C" void kernel_launch(void* const* d_in, const int* in_sizes, int n_in,
                              void* d_out, int out_size, void* d_ws, size_t ws_size,
                              hipStream_t stream) {
    (void)in_sizes; (void)n_in; (void)out_size; (void)ws_size;

    const float* x      = (const float*)d_in[0];   // [N,128]
    const float* W_emb  = (const float*)d_in[1];   // [128,128]
    const float* b_emb  = (const float*)d_in[2];   // [128]
    const float* Wc     = (const float*)d_in[3];   // [L,128,128]
    const float* bc     = (const float*)d_in[4];   // [L,128]
    const float* gamma  = (const float*)d_in[5];   // [L,128]
    const float* beta   = (const float*)d_in[6];   // [L,128]
    const float* W1     = (const float*)d_in[7];   // [128,64]
    const float* b1     = (const float*)d_in[8];   // [64]
    const float* W2     = (const float*)d_in[9];   // [64,32]
    const float* b2     = (const float*)d_in[10];  // [32]
    const float* W3     = (const float*)d_in[11];  // [32,10]
    const float* b3     = (const float*)d_in[12];  // [10]
    const int*   ei     = (const int*)d_in[13];    // [2,E]
    const int*   batch  = (const int*)d_in[14];    // [N]
    float*       out    = (float*)d_out;           // [G,T]

    const int* src = ei;
    const int* dst = ei + EE;

    // Workspace layout (floats)
    float* ws     = (float*)d_ws;
    float* h      = ws;                  // N*D
    float* hw     = h + (size_t)NN * DD; // N*D
    float* agg    = hw + (size_t)NN * DD;// N*D
    float* dinv   = agg + (size_t)NN * DD; // N
    float* colsum = dinv + NN;           // 128
    float* colsq  = colsum + DD;         // 128
    float* mu     = colsq + DD;          // 128
    float* scale  = mu + DD;             // 128
    float* cnt    = scale + DD;          // G
    float* pooled = cnt + GG;            // G*D  (contiguous after cnt)

    const int ntiles     = NN / 16;                       // 6250 row tiles
    const int gemm_blks  = (ntiles + 7) / 8;              // 8 waves / 256-thread block
    const int nd         = NN * DD;
    const int nd_blks    = (nd + 255) / 256;
    const int edge_blks  = (EE * 32 + 255) / 256;         // wave per edge
    const int node_blks  = (NN * 32 + 255) / 256;         // wave per node

    // Degree -> D^{-1/2}
    qgcn_deg_init<<<(NN + 255) / 256, 256, 0, stream>>>(dinv, NN);
    qgcn_deg_edges<<<(EE + 255) / 256, 256, 0, stream>>>(dst, dinv, EE);
    qgcn_deg_finalize<<<(NN + 255) / 256, 256, 0, stream>>>(dinv, NN);

    // h = x @ W_emb + b_emb
    qgcn_gemm_wmma<<<gemm_blks, 256, 0, stream>>>(x, W_emb, b_emb, h, ntiles);

    for (int l = 0; l < LL; ++l) {
        // hw = h @ Wc[l]
        qgcn_gemm_wmma<<<gemm_blks, 256, 0, stream>>>(h, Wc + (size_t)l * DD * DD,
                                                      nullptr, hw, ntiles);
        // agg = bc[l] + self-loop message
        qgcn_selfloop_init<<<nd_blks, 256, 0, stream>>>(hw, dinv, bc + l * DD, agg, nd);
        // scatter edge messages
        qgcn_edge_scatter<<<edge_blks, 256, 0, stream>>>(src, dst, dinv, hw, agg, EE);
        // BN stats
        qgcn_zero<<<1, 256, 0, stream>>>(colsum, 2 * DD);   // colsum + colsq contiguous
        qgcn_bn_stats<<<512, 256, 0, stream>>>(agg, colsum, colsq, NN);
        qgcn_bn_finalize<<<1, 128, 0, stream>>>(colsum, colsq, gamma + l * DD, mu, scale);
        // h += relu(bn(agg))
        qgcn_bn_update<<<nd_blks, 256, 0, stream>>>(h, agg, mu, scale, beta + l * DD, nd);
    }

    // Pool (cnt + pooled contiguous: G + G*D floats)
    qgcn_zero<<<(GG + GG * DD + 255) / 256, 256, 0, stream>>>(cnt, GG + GG * DD);
    qgcn_pool_sum<<<node_blks, 256, 0, stream>>>(h, batch, pooled, cnt, NN);

    // Readout MLP
    qgcn_mlp<<<GG, 128, 0, stream>>>(pooled, cnt, W1, b1, W2, b2, W3, b3, out);
}